// SampleBoxLoss_70480413328153
// MI455X (gfx1250) — compile-verified
//
#include <hip/hip_runtime.h>
#include <hip/hip_bf16.h>

typedef __attribute__((ext_vector_type(2))) float v2f;
typedef __attribute__((ext_vector_type(8))) float v8f;

#define B_ 4
#define N_ 16384
#define M_ 2048
#define ROWS_PER_BLOCK 128
#define THREADS 256
#define RT 512

// ---------------------------------------------------------------------------
// Kernel 0: init global column-min accumulator to +inf bits
// ---------------------------------------------------------------------------
__global__ void sbl_init_colmin(unsigned* __restrict__ colmin) {
    int i = blockIdx.x * blockDim.x + threadIdx.x;
    if (i < B_ * M_) colmin[i] = 0x7F800000u;  // +inf
}

// ---------------------------------------------------------------------------
// Kernel 1: pairwise d2 via V_WMMA_F32_16X16X4_F32 + dual min reductions
//   d2[n,m] = |x_n|^2 + ( A(x_n,..,1) x B(-2p_m,..,|p_m|^2) )
// Block: 256 thr (8 waves). Each wave owns 16 N-rows; block owns 128 rows
// and sweeps all M=2048 columns with the surface matrix resident in LDS.
// ---------------------------------------------------------------------------
__global__ __launch_bounds__(THREADS) void sbl_pairwise_min(
    const float* __restrict__ lidar,    // (B,N,4) f32
    const float* __restrict__ surf,     // (B,M,3) f32
    float*       __restrict__ rowmin,   // (B,N)   raw min d2 (unclamped)
    unsigned*    __restrict__ colmin)   // (B,M)   f32 bits of clamped min d2
{
    __shared__ float    sB[M_ * 4];     // 32KB: (-2x,-2y,-2z,|p|^2) per point
    __shared__ unsigned sCol[M_];       // 8KB : per-block column mins

    const int b   = blockIdx.y;
    const int tid = threadIdx.x;

    // --- build B operand in LDS ---
    const float* sp = surf + (size_t)b * M_ * 3;
    for (int i = tid; i < M_; i += THREADS) {
        float px = sp[3 * i + 0];
        float py = sp[3 * i + 1];
        float pz = sp[3 * i + 2];
        sB[4 * i + 0] = -2.0f * px;
        sB[4 * i + 1] = -2.0f * py;
        sB[4 * i + 2] = -2.0f * pz;
        sB[4 * i + 3] = px * px + py * py + pz * pz;
        sCol[i]       = 0x7F800000u;
    }
    __syncthreads();

    const int wave = tid >> 5;
    const int lane = tid & 31;
    const int kh   = lane >> 4;   // 0: K={0,1}, rows 0-7 of C ; 1: K={2,3}, rows 8-15
    const int r    = lane & 15;   // A: row index for load; B/C: column index
    const int n0   = blockIdx.x * ROWS_PER_BLOCK + wave * 16;

    // A operand: lane<16 -> (x,y), lane>=16 -> (z,1); lidar stride-4 is float4
    const float4 p = *(const float4*)(lidar + ((size_t)b * N_ + n0 + r) * 4);
    const float  x2 = p.x * p.x + p.y * p.y + p.z * p.z;
    v2f a;
    a.x = kh ? p.z : p.x;
    a.y = kh ? 1.0f : p.y;

    // per-lane |x|^2 for the 8 C/D rows this lane produces (rows kh*8 + v)
    float x2r[8];
#pragma unroll
    for (int v = 0; v < 8; ++v) x2r[v] = __shfl(x2, kh * 8 + v, 32);

    float racc[8];
#pragma unroll
    for (int v = 0; v < 8; ++v) racc[v] = 3.4e38f;

    const v2f* bpairs = (const v2f*)sB;  // pair index = m*2 + kh

#pragma unroll 2
    for (int mt = 0; mt < M_ / 16; ++mt) {
        const int m = mt * 16 + r;
        v2f bv = bpairs[m * 2 + kh];
        v8f c  = {};
        c = __builtin_amdgcn_wmma_f32_16x16x4_f32(
                /*neg_a=*/false, a, /*neg_b=*/false, bv,
                /*c_mod=*/(short)0, c, /*reuse_a=*/false, /*reuse_b=*/false);
        float cm = 3.4e38f;
#pragma unroll
        for (int v = 0; v < 8; ++v) {
            float d2 = c[v] + x2r[v];            // raw squared distance
            racc[v]  = fminf(racc[v], d2);       // row (over-M) min, unclamped
            cm       = fminf(cm, d2);            // column (over-N) partial
        }
        cm = fmaxf(cm, 0.0f);                    // clamp once -> uint-orderable
        atomicMin(&sCol[m], __float_as_uint(cm));
    }

    // row min across the 16 lanes of each half (XOR tree stays within half)
#pragma unroll
    for (int off = 1; off < 16; off <<= 1) {
#pragma unroll
        for (int v = 0; v < 8; ++v)
            racc[v] = fminf(racc[v], __shfl_xor(racc[v], off, 32));
    }
    if (r == 0) {
        float* out = rowmin + (size_t)b * N_ + n0 + kh * 8;
#pragma unroll
        for (int v = 0; v < 8; ++v) out[v] = racc[v];
    }

    __syncthreads();
    for (int i = tid; i < M_; i += THREADS)
        atomicMin(&colmin[(size_t)b * M_ + i], sCol[i]);
}

// ---------------------------------------------------------------------------
// Kernel 2: final scalar — sqrt deferred here (monotone under min/max)
//   LB=mean_m sqrt(colmin), LF=mean_n sqrt(rowmin), LM=max_n sqrt(rowmin)
// ---------------------------------------------------------------------------
__global__ __launch_bounds__(RT) void sbl_final_reduce(
    const float* __restrict__ rowmin, const unsigned* __restrict__ colmin,
    float* __restrict__ out)
{
    __shared__ float sLF[RT], sLM[RT], sLB[RT];
    const int tid = threadIdx.x;
    float total = 0.0f;

    for (int b = 0; b < B_; ++b) {
        float lf = 0.0f, lm = 0.0f, lb = 0.0f;
        for (int i = tid; i < N_; i += RT) {
            float v = sqrtf(fmaxf(rowmin[b * N_ + i], 0.0f));
            lf += v;
            lm = fmaxf(lm, v);
        }
        for (int i = tid; i < M_; i += RT)
            lb += sqrtf(fmaxf(__uint_as_float(colmin[b * M_ + i]), 0.0f));

        sLF[tid] = lf; sLM[tid] = lm; sLB[tid] = lb;
        __syncthreads();
        for (int s = RT / 2; s > 0; s >>= 1) {
            if (tid < s) {
                sLF[tid] += sLF[tid + s];
                sLM[tid]  = fmaxf(sLM[tid], sLM[tid + s]);
                sLB[tid] += sLB[tid + s];
            }
            __syncthreads();
        }
        if (tid == 0)
            total += 5.0f * (sLB[0] / (float)M_) + sLF[0] / (float)N_ + sLM[0];
        __syncthreads();
    }
    if (tid == 0) out[0] = total / (float)B_;
}

// ---------------------------------------------------------------------------
extern "C" void kernel_launch(void* const* d_in, const int* in_sizes, int n_in,
                              void* d_out, int out_size, void* d_ws, size_t ws_size,
                              hipStream_t stream) {
    (void)in_sizes; (void)n_in; (void)out_size; (void)ws_size;
    const float* lidar = (const float*)d_in[0];   // (B,N,4)
    const float* surf  = (const float*)d_in[1];   // (B,M,3)

    float*    rowmin = (float*)d_ws;                       // B*N floats
    unsigned* colmin = (unsigned*)(rowmin + B_ * N_);      // B*M uints

    sbl_init_colmin<<<(B_ * M_ + 255) / 256, 256, 0, stream>>>(colmin);

    dim3 grid(N_ / ROWS_PER_BLOCK, B_);
    sbl_pairwise_min<<<grid, THREADS, 0, stream>>>(lidar, surf, rowmin, colmin);

    sbl_final_reduce<<<1, RT, 0, stream>>>(rowmin, colmin, (float*)d_out);
}